// S4_18683107738068
// MI455X (gfx1250) — compile-verified
//
#include <hip/hip_runtime.h>
#include <math.h>

// ---------------------------------------------------------------------------
// S4 DPLR convolution for MI455X (gfx1250, wave32).
//   out[b,t] = (u[b] * K)[t] + d*u[b,t],  K = DPLR SSM kernel (L=16384)
// FFT convolution. The 16384-pt complex FFT is a four-step 128x128 algorithm
// whose radix-128 DFTs are chained V_WMMA_F32_16X16X4_F32 ops (full fp32).
// One WG per row, 512 threads (16 waves = 4/SIMD), ~299KB of 320KB WGP LDS.
// - Inverse transforms via ifft(x)=conj(fft(conj(x)))/N (always-forward WMMA).
// - d*u folded into the frequency kernel (Kd += d), so u is read exactly once
//   (async-staged into LDS via global_load_async_to_lds_b128 / ASYNCcnt) and
//   HBM traffic is the 256MB minimum (~11us floor at 23.3 TB/s).
// - GEMM1 output stored in a row-pair interleaved layout (stride 288 = 32 mod
//   64 banks) so GEMM2 B-fragments are single conflict-free ds_load_b64s.
// ---------------------------------------------------------------------------

typedef float v2f __attribute__((ext_vector_type(2)));
typedef float v8f __attribute__((ext_vector_type(8)));

#define L_LEN     16384
#define NROWS     2048
#define NSTATE    64
#define NTHREADS  512
#define PL        132               // src planes: row-major, padded stride
#define TP        288               // tmp planes: row-pair layout stride
#define PSZ       18432             // plane size (floats) = max(128*132,64*288)
#define LDS_FLOATS (4 * PSZ + 1024) // 4 planes + 4 root tables (128 x v2f)
#define LDS_BYTES  (LDS_FLOATS * sizeof(float))
#define STAGE_BYTE_OFF (2 * PSZ * 4) // u-row staging aliases the P1re plane
#define INV_N     (1.0f / 16384.0f)
#define TWO_PI    6.283185307179586f

static __device__ __forceinline__ v2f cmul(v2f a, v2f b) {
  v2f r; r.x = a.x * b.x - a.y * b.y; r.y = a.x * b.y + a.y * b.x; return r;
}

// D = A(16x4) x B(4x16) + C, fp32 WMMA.
static __device__ __forceinline__ v8f wmma4(v2f a, v2f b, v8f c) {
  return __builtin_amdgcn_wmma_f32_16x16x4_f32(false, a, false, b, (short)0, c,
                                               false, false);
}

// ---------------------------------------------------------------------------
// In-LDS 16384-pt complex forward DFT (four-step, N1=N2=128).
// Input  x[n] at src[n&127][n>>7]   (row-major PL layout, n = n1 + 128*n2)
// Output X[k] at src[k>>7][k&127]   (row-major PL layout, k = k2 + 128*k1)
// tmp planes (row-pair TP layout) are clobbered. Barrier before entry.
// kmax: K extent of GEMM1 (64 when input cols n2>=64 are known zero).
// Wave w (of 16) owns tile-row w>>1, column tiles (w&1)*4 .. +3.
// ---------------------------------------------------------------------------
static __device__ __forceinline__ void gemm_fft(
    float* __restrict__ sRe, float* __restrict__ sIm,
    float* __restrict__ tRe, float* __restrict__ tIm,
    const v2f* __restrict__ w128, const v2f* __restrict__ w16k,
    int kmax, int tid)
{
  const int wv   = tid >> 5;
  const int lane = tid & 31;
  const int m    = lane & 15;        // A row / B col within tile
  const int kh   = lane >> 4;        // K half for 16x16x4 fragments
  const int r0   = (wv >> 1) * 16;   // tile-row base
  const int c0   = (wv & 1) * 4;     // first column tile

  // ---- GEMM1: T[n1][k2] = sum_n2 src[n1][n2] * W128^{n2*k2}, then twiddle.
  {
    v8f accRe[4] = {}, accIm[4] = {};
    const float* rowRe = sRe + (r0 + m) * PL;
    const float* rowIm = sIm + (r0 + m) * PL;
    for (int k0 = 0; k0 < kmax; k0 += 4) {
      const int kk = k0 + 2 * kh;
      v2f aRe  = *(const v2f*)(rowRe + kk);
      v2f aIm  = *(const v2f*)(rowIm + kk);
      v2f aImN = -aIm;                       // one negation per K-step
#pragma unroll
      for (int c = 0; c < 4; ++c) {
        const int col = (c0 + c) * 16 + m;
        v2f t0 = w128[(kk * col) & 127];
        v2f t1 = w128[((kk + 1) * col) & 127];
        v2f bRe; bRe.x = t0.x; bRe.y = t1.x;
        v2f bIm; bIm.x = t0.y; bIm.y = t1.y;
        accRe[c] = wmma4(aRe,  bRe, accRe[c]);
        accRe[c] = wmma4(aImN, bIm, accRe[c]);   // -Mim*Fim
        accIm[c] = wmma4(aRe,  bIm, accIm[c]);
        accIm[c] = wmma4(aIm,  bRe, accIm[c]);
      }
    }
    // twiddle by W_16384^{n1*k2} = W128^{idx>>7} * W16384^{idx&127};
    // store into the row-pair layout: addr = (row>>1)*TP + 2*col + (row&1)
#pragma unroll
    for (int c = 0; c < 4; ++c) {
      const int col = (c0 + c) * 16 + m;
#pragma unroll
      for (int v = 0; v < 8; ++v) {
        const int row = r0 + v + 8 * kh;     // n1 per C/D VGPR layout
        const int idx = (row * col) & 16383;
        v2f wq = w128[idx >> 7];
        v2f wr = w16k[idx & 127];
        float twre = wq.x * wr.x - wq.y * wr.y;
        float twim = wq.x * wr.y + wq.y * wr.x;
        float xr = accRe[c][v], xi = accIm[c][v];
        const int pa = (row >> 1) * TP + 2 * col + (row & 1);
        tRe[pa] = xr * twre - xi * twim;
        tIm[pa] = xr * twim + xi * twre;
      }
    }
  }
  __syncthreads();

  // ---- GEMM2: Out[k1][k2] = sum_n1 W128^{k1*n1} * T'[n1][k2] -> src planes.
  {
    v8f accRe[4] = {}, accIm[4] = {};
    for (int k0 = 0; k0 < 128; k0 += 4) {
      const int kk = k0 + 2 * kh;
      v2f t0 = w128[((r0 + m) * kk) & 127];
      v2f t1 = w128[((r0 + m) * (kk + 1)) & 127];
      v2f aRe; aRe.x = t0.x; aRe.y = t1.x;
      v2f aIm; aIm.x = t0.y; aIm.y = t1.y;
      v2f aImN = -aIm;
      const int pr = (kk >> 1) * TP;         // row-pair base for {kk, kk+1}
#pragma unroll
      for (int c = 0; c < 4; ++c) {
        const int col = (c0 + c) * 16 + m;
        v2f bRe = *(const v2f*)(tRe + pr + 2 * col);  // {T[kk][col],T[kk+1][col]}
        v2f bIm = *(const v2f*)(tIm + pr + 2 * col);
        accRe[c] = wmma4(aRe,  bRe, accRe[c]);
        accRe[c] = wmma4(aImN, bIm, accRe[c]);
        accIm[c] = wmma4(aRe,  bIm, accIm[c]);
        accIm[c] = wmma4(aIm,  bRe, accIm[c]);
      }
    }
#pragma unroll
    for (int c = 0; c < 4; ++c) {
      const int col = (c0 + c) * 16 + m;
#pragma unroll
      for (int v = 0; v < 8; ++v) {
        const int row = r0 + v + 8 * kh;     // k1
        sRe[row * PL + col] = accRe[c][v];
        sIm[row * PL + col] = accIm[c][v];
      }
    }
  }
  __syncthreads();
}

// ---------------------------------------------------------------------------
// Cauchy / Woodbury: at_roots[l] for l in [0,16384)
// ---------------------------------------------------------------------------
__global__ void s4_atroots_kernel(
    const float* __restrict__ lre, const float* __restrict__ lim,
    const float* __restrict__ pv,  const float* __restrict__ bv,
    const float* __restrict__ cv,  const float* __restrict__ delta,
    v2f* __restrict__ atroots)
{
  __shared__ float sLr[NSTATE], sLi[NSTATE], sP[NSTATE], sB[NSTATE],
                   sC0[NSTATE], sC1[NSTATE];
  const int tid = threadIdx.x;
  if (tid < NSTATE) {
    sLr[tid] = fminf(lre[tid], -1e-4f);
    sLi[tid] = lim[tid];
    sP[tid]  = pv[tid];
    sB[tid]  = bv[tid];
    sC0[tid] = cv[2 * tid];
    sC1[tid] = cv[2 * tid + 1];
  }
  __syncthreads();
  const int l = blockIdx.x * blockDim.x + tid;
  if (l >= L_LEN) return;

  const float step = expf(delta[0]);
  float sn, cn;
  sincosf(-TWO_PI * (float)l / (float)L_LEN, &sn, &cn);  // Omega_L^l
  v2f onep; onep.x = 1.f + cn; onep.y = sn;
  v2f onem; onem.x = 1.f - cn; onem.y = -sn;
  const float den = onep.x * onep.x + onep.y * onep.y;
  const float sc  = 2.f / step;
  v2f g;  g.x  = sc * (onem.x * onep.x + onem.y * onep.y) / den;
          g.y  = sc * (onem.y * onep.x - onem.x * onep.y) / den;
  v2f c2; c2.x = 2.f * onep.x / den; c2.y = -2.f * onep.y / den;

  v2f k00 = {}, k01 = {}, k10 = {}, k11 = {};
  for (int n = 0; n < NSTATE; ++n) {
    v2f dd; dd.x = g.x - sLr[n]; dd.y = g.y - sLi[n];
    const float id = 1.f / (dd.x * dd.x + dd.y * dd.y);
    v2f inv; inv.x = dd.x * id; inv.y = -dd.y * id;       // 1/(g - Lambda)
    v2f a0;  a0.x = sC0[n];     a0.y = -sC1[n];           // conj(C)
    v2f e = cmul(a0, inv);
    const float pn = sP[n], bn = sB[n];
    k00.x += bn * e.x;   k00.y += bn * e.y;
    k01.x += pn * e.x;   k01.y += pn * e.y;
    const float pb = pn * bn;
    k10.x += pb * inv.x; k10.y += pb * inv.y;
    const float pp = pn * pn;
    k11.x += pp * inv.x; k11.y += pp * inv.y;
  }
  v2f dn; dn.x = 1.f + k11.x; dn.y = k11.y;
  v2f num = cmul(k01, k10);
  const float dd2 = dn.x * dn.x + dn.y * dn.y;
  v2f q; q.x = (num.x * dn.x + num.y * dn.y) / dd2;
         q.y = (num.y * dn.x - num.x * dn.y) / dd2;
  v2f t; t.x = k00.x - q.x; t.y = k00.y - q.y;
  atroots[l] = cmul(c2, t);
}

// ---------------------------------------------------------------------------
// Workgroup FFT kernel. mode 0: K = Re(ifft(at_roots))        (1 WG)
//                       mode 1: Kd = rfft(K, 32768) + d       (1 WG)
//                       mode 2: per-row FFT convolution       (2048 WGs)
// ---------------------------------------------------------------------------
__global__ __launch_bounds__(NTHREADS, 1) void s4_fft_kernel(
    int mode,
    const float* __restrict__ u, const float* __restrict__ dcoef,
    float* __restrict__ out,
    const v2f* __restrict__ atroots, float* __restrict__ Kw,
    v2f* __restrict__ Kd)
{
  extern __shared__ char dynsmem[];
  float* lds  = (float*)dynsmem;
  float* P0re = lds;
  float* P0im = lds + PSZ;
  float* P1re = lds + 2 * PSZ;
  float* P1im = lds + 3 * PSZ;
  v2f*   w128 = (v2f*)(lds + 4 * PSZ);  // e^{-2pi i t/128}
  v2f*   w16k = w128 + 128;             // e^{-2pi i t/16384}
  v2f*   w256 = w16k + 128;             // e^{-2pi i t/256}
  v2f*   w32k = w256 + 128;             // e^{-2pi i t/32768}
  const int tid = threadIdx.x;

  if (tid < 128) {
    v2f a; a.x = cosf(-TWO_PI * (float)tid / 128.0f);
           a.y = sinf(-TWO_PI * (float)tid / 128.0f);
    v2f b; b.x = cosf(-TWO_PI * (float)tid / 16384.0f);
           b.y = sinf(-TWO_PI * (float)tid / 16384.0f);
    v2f c; c.x = cosf(-TWO_PI * (float)tid / 256.0f);
           c.y = sinf(-TWO_PI * (float)tid / 256.0f);
    v2f d; d.x = cosf(-TWO_PI * (float)tid / 32768.0f);
           d.y = sinf(-TWO_PI * (float)tid / 32768.0f);
    w128[tid] = a; w16k[tid] = b; w256[tid] = c; w32k[tid] = d;
  }

  if (mode == 0) {  // K = Re(ifft(at_roots)) = Re(fft(conj(at_roots)))/N
    __syncthreads();
    for (int n = tid; n < L_LEN; n += NTHREADS) {
      v2f z = atroots[n];
      P0re[(n & 127) * PL + (n >> 7)] = z.x;
      P0im[(n & 127) * PL + (n >> 7)] = -z.y;   // conj
    }
    __syncthreads();
    gemm_fft(P0re, P0im, P1re, P1im, w128, w16k, 128, tid);
    for (int t = tid; t < L_LEN; t += NTHREADS)
      Kw[t] = P0re[(t >> 7) * PL + (t & 127)] * INV_N;
    return;
  }

  if (mode == 1) {  // Kd = rfft(K, 2L) + d: pack even/odd, fwd FFT, unpack
    const float dval = dcoef[0];
    __syncthreads();
    for (int n = tid; n < 8192; n += NTHREADS) {  // top half zero: skipped
      v2f kk = ((const v2f*)Kw)[n];
      P0re[(n & 127) * PL + (n >> 7)] = kk.x;
      P0im[(n & 127) * PL + (n >> 7)] = kk.y;
    }
    __syncthreads();
    gemm_fft(P0re, P0im, P1re, P1im, w128, w16k, 64, tid);
    for (int j = tid; j <= 8192; j += NTHREADS) {
      const int k = j, mk = L_LEN - j, mz = mk & (L_LEN - 1);
      v2f Zk; Zk.x = P0re[(k  >> 7) * PL + (k  & 127)];
              Zk.y = P0im[(k  >> 7) * PL + (k  & 127)];
      v2f Zm; Zm.x = P0re[(mz >> 7) * PL + (mz & 127)];
              Zm.y = P0im[(mz >> 7) * PL + (mz & 127)];
      v2f Ze; Ze.x = 0.5f * (Zk.x + Zm.x); Ze.y = 0.5f * (Zk.y - Zm.y);
      v2f Zo; Zo.x = 0.5f * (Zk.y + Zm.y); Zo.y = -0.5f * (Zk.x - Zm.x);
      v2f W = cmul(w256[k >> 7], w32k[k & 127]);  // e^{-i pi k/16384}
      v2f WZo = cmul(W, Zo);
      v2f Ud;  Ud.x  = Ze.x + WZo.x + dval; Ud.y  = Ze.y + WZo.y;
      v2f Udm; Udm.x = Ze.x - WZo.x + dval; Udm.y = -(Ze.y - WZo.y);
      Kd[k]  = Ud;    // rfft(K,2L)[k] + d  (d*u folded into the kernel)
      Kd[mk] = Udm;
    }
    return;
  }

  // ---- mode 2: one row's causal convolution -------------------------------
  const int row = blockIdx.x;
  const float* urow = u + (size_t)row * L_LEN;

  // Async-stage the 64KB u row into LDS (aliases P1re, free until GEMM1).
  // Dynamic LDS starts at offset 0 here (no static LDS in this kernel).
  {
    const char* g = (const char*)urow;
#pragma unroll
    for (int i = 0; i < 8; ++i) {
      const unsigned boff = (unsigned)((i * NTHREADS + tid) * 16);
      asm volatile("global_load_async_to_lds_b128 %0, %1, off"
                   :: "v"(STAGE_BYTE_OFF + boff), "v"(g + boff) : "memory");
    }
    asm volatile("s_wait_asynccnt 0x0" ::: "memory");
  }
  __syncthreads();
  {
    const v2f* ustage = (const v2f*)P1re;     // z[n] = u[2n] + i*u[2n+1]
    for (int n = tid; n < 8192; n += NTHREADS) {
      v2f t = ustage[n];
      P0re[(n & 127) * PL + (n >> 7)] = t.x;
      P0im[(n & 127) * PL + (n >> 7)] = t.y;
    }
  }
  __syncthreads();
  gemm_fft(P0re, P0im, P1re, P1im, w128, w16k, 64, tid);

  // unpack -> Ud, multiply by (Kd+d), repack -> conj(Z')/N (transposed)
  for (int j = tid; j <= 8192; j += NTHREADS) {
    const int k = j, mk = L_LEN - j, mz = mk & (L_LEN - 1);
    v2f Zk; Zk.x = P0re[(k  >> 7) * PL + (k  & 127)];
            Zk.y = P0im[(k  >> 7) * PL + (k  & 127)];
    v2f Zm; Zm.x = P0re[(mz >> 7) * PL + (mz & 127)];
            Zm.y = P0im[(mz >> 7) * PL + (mz & 127)];
    v2f Ze; Ze.x = 0.5f * (Zk.x + Zm.x); Ze.y = 0.5f * (Zk.y - Zm.y);
    v2f Zo; Zo.x = 0.5f * (Zk.y + Zm.y); Zo.y = -0.5f * (Zk.x - Zm.x);
    v2f W = cmul(w256[k >> 7], w32k[k & 127]);    // e^{-i pi k/16384}
    v2f WZo = cmul(W, Zo);
    v2f Ud;  Ud.x = Ze.x + WZo.x;  Ud.y = Ze.y + WZo.y;
    v2f Udm; Udm.x = Ze.x - WZo.x; Udm.y = -(Ze.y - WZo.y);   // conj(Ze-WZo)
    v2f Yk = cmul(Ud,  Kd[k]);
    v2f Ym = cmul(Udm, Kd[mk]);
    v2f Ye; Ye.x = 0.5f * (Yk.x + Ym.x); Ye.y = 0.5f * (Yk.y - Ym.y);
    v2f Yo; Yo.x = 0.5f * (Yk.x - Ym.x); Yo.y = 0.5f * (Yk.y + Ym.y);
    v2f V; V.x = W.x; V.y = -W.y;                 // e^{+i pi k/16384}
    v2f VY = cmul(V, Yo);
    // Z'[k]/N, stored conjugated for the conj-trick inverse
    float zkx = (Ye.x - VY.y) * INV_N, zky = (Ye.y + VY.x) * INV_N;
    v2f Yoc; Yoc.x = Yo.x; Yoc.y = -Yo.y;
    v2f WY = cmul(W, Yoc);
    float zmx = (Ye.x - WY.y) * INV_N, zmy = (-Ye.y + WY.x) * INV_N;
    P1re[(k  & 127) * PL + (k  >> 7)] = zkx;
    P1im[(k  & 127) * PL + (k  >> 7)] = -zky;
    P1re[(mz & 127) * PL + (mz >> 7)] = zmx;
    P1im[(mz & 127) * PL + (mz >> 7)] = -zmy;
  }
  __syncthreads();
  gemm_fft(P1re, P1im, P0re, P0im, w128, w16k, 128, tid);

  // result R = conj(z'); out[2n] = Re R, out[2n+1] = -Im R (d*u already in Kd)
  v2f* out2 = (v2f*)(out + (size_t)row * L_LEN);
  for (int n = tid; n < 8192; n += NTHREADS) {
    v2f o;
    o.x =  P1re[(n >> 7) * PL + (n & 127)];
    o.y = -P1im[(n >> 7) * PL + (n & 127)];
    out2[n] = o;
  }
}

// ---------------------------------------------------------------------------
extern "C" void kernel_launch(void* const* d_in, const int* in_sizes, int n_in,
                              void* d_out, int out_size, void* d_ws,
                              size_t ws_size, hipStream_t stream)
{
  (void)in_sizes; (void)n_in; (void)out_size; (void)ws_size;
  const float* u     = (const float*)d_in[0];
  const float* lre   = (const float*)d_in[1];
  const float* lim   = (const float*)d_in[2];
  const float* pv    = (const float*)d_in[3];
  const float* bv    = (const float*)d_in[4];
  const float* cv    = (const float*)d_in[5];
  const float* dv    = (const float*)d_in[6];
  const float* delta = (const float*)d_in[7];
  float* out = (float*)d_out;

  char* ws = (char*)d_ws;
  v2f*   atroots = (v2f*)ws;                        // 16384 * 8B
  float* Kw      = (float*)(ws + 131072);           // 16384 * 4B
  v2f*   Kd      = (v2f*)(ws + 131072 + 65536);     // 16385 * 8B

  s4_atroots_kernel<<<L_LEN / 256, 256, 0, stream>>>(lre, lim, pv, bv, cv,
                                                     delta, atroots);
  s4_fft_kernel<<<1, NTHREADS, LDS_BYTES, stream>>>(0, nullptr, nullptr,
                                                    nullptr, atroots, Kw, Kd);
  s4_fft_kernel<<<1, NTHREADS, LDS_BYTES, stream>>>(1, nullptr, dv,
                                                    nullptr, atroots, Kw, Kd);
  s4_fft_kernel<<<NROWS, NTHREADS, LDS_BYTES, stream>>>(2, u, dv, out,
                                                        atroots, Kw, Kd);
}